// GMM_73272142069782
// MI455X (gfx1250) — compile-verified
//
#include <hip/hip_runtime.h>
#include <math.h>

typedef __attribute__((ext_vector_type(2))) float v2f;
typedef __attribute__((ext_vector_type(4))) float v4f;
typedef __attribute__((ext_vector_type(8))) float v8f;

#define KCL 15
#define MW 100
#define SEDGE 120
#define SM (SEDGE * MW)              // 12000
#define SM_TILES_PAD 768             // 750 real 16-col tiles, padded to 768 (96 per wave)
#define SM_PAD (SM_TILES_PAD * 16)   // 12288
#define NB 4096
#define C0F (-58.81206612f)          // -0.5 * 64 * log(2*pi)
#define NEG_BIG (-1.0e30f)

// workspace layout (in floats)
#define WS_LOGPI 0
#define WS_Q 128
#define WS_BFRAG (WS_Q + SM_PAD)                       // 768 tiles * 1024 floats
#define WS_PART (WS_BFRAG + SM_TILES_PAD * 1024)       // 256 block partials

// ---------------------------------------------------------------- kernel 0
// log(softmax(pi) + 1e-30) -> ws[WS_LOGPI .. +120]
__global__ void gmm_logpi_kernel(const float* __restrict__ pi, float* __restrict__ wsf) {
    __shared__ float buf[128];
    const int t = threadIdx.x;
    float x = (t < SEDGE) ? pi[t] : -3.0e38f;
    buf[t] = x;
    __syncthreads();
    for (int s = 64; s > 0; s >>= 1) { if (t < s) buf[t] = fmaxf(buf[t], buf[t + s]); __syncthreads(); }
    float mx = buf[0];
    __syncthreads();
    buf[t] = (t < SEDGE) ? __expf(x - mx) : 0.0f;
    __syncthreads();
    for (int s = 64; s > 0; s >>= 1) { if (t < s) buf[t] += buf[t + s]; __syncthreads(); }
    float lse = mx + __logf(buf[0]);
    if (t < SEDGE) {
        float ls = x - lse;
        wsf[WS_LOGPI + t] = __logf(__expf(ls) + 1e-30f);
    }
}

// ---------------------------------------------------------------- kernel 1
// Build mu_sm in WMMA-B fragment layout (two K-groups packed per 16B quad so the
// main loop uses global_load_b128) + q[sm] = log_pi[s] - 0.5*||mu_sm||^2.
// For K-index d, column c of tile j:
//   g = d>>2 (K-group), p = g>>1 (quad pair), gi = g&1, r = d&3
//   lane = (r>=2 ? 16 : 0) + c, v = r&1
//   float index = j*1024 + p*128 + lane*4 + gi*2 + v
__global__ void gmm_prep_kernel(const float* __restrict__ mu, float* __restrict__ wsf) {
    const int j = blockIdx.x;
    const int t = threadIdx.x;
    const int c = t & 15;        // column in tile
    const int dg = t >> 4;       // 0..15, handles K-indices dg*4 .. dg*4+3
    __shared__ float mmacc[16];
    if (t < 16) mmacc[t] = 0.0f;
    __syncthreads();

    const int sm = j * 16 + c;
    float vals[4];
    float ss = 0.0f;
    if (sm < SM) {
        const int s = sm / MW;
        const int m = sm - s * MW;
        int rem = s, a = 0;
        while (rem >= KCL - a) { rem -= KCL - a; ++a; }   // upper-tri (a,b) decode
        const int b = a + rem;
        const float w = (float)m / (float)MW;
        #pragma unroll
        for (int i = 0; i < 4; ++i) {
            const int d = dg * 4 + i;
            const float v = mu[d * KCL + a] * w + mu[d * KCL + b] * (1.0f - w);
            vals[i] = v;
            ss += v * v;
        }
    } else {
        vals[0] = vals[1] = vals[2] = vals[3] = 0.0f;
    }

    float* bf = wsf + WS_BFRAG;
    const int g = dg;                 // this thread's K-group
    const int p = g >> 1, gi = g & 1;
    #pragma unroll
    for (int i = 0; i < 4; ++i) {
        const int r = i;              // d = g*4 + r
        const int lane = ((r >> 1) << 4) + c;
        const int v = r & 1;
        bf[(size_t)j * 1024 + p * 128 + lane * 4 + gi * 2 + v] = vals[i];
    }
    atomicAdd(&mmacc[c], ss);
    __syncthreads();
    if (t < 16) {
        const int sm2 = j * 16 + t;
        float q;
        if (sm2 < SM) { const int s2 = sm2 / MW; q = wsf[WS_LOGPI + s2] - 0.5f * mmacc[t]; }
        else          { q = NEG_BIG; }
        wsf[WS_Q + sm2] = q;
    }
}

// ---------------------------------------------------------------- kernel 2
// Fused GEMM (V_WMMA_F32_16X16X4_F32, dual accumulator chains) + per-lane online
// logsumexp; lanes butterfly-merge (m,s) once after the 96-tile sweep.
__global__ void __launch_bounds__(256, 2)
gmm_main_kernel(const float* __restrict__ z, const float* __restrict__ wsq,
                const float* __restrict__ wsbf, float* __restrict__ partial) {
    __shared__ float zlds[16 * 68];   // padded stride 68 -> conflict-free ds_load_b64
    __shared__ float zz[16];
    __shared__ float red_m[8 * 16];
    __shared__ float red_s[8 * 16];
    __shared__ float rowlp[16];

    const int t = threadIdx.x;
    const int lane = t & 31;
    const int wave = t >> 5;
    const int row0 = blockIdx.x * 16;

    // stage z tile [16][64] into LDS
    #pragma unroll
    for (int i = 0; i < 4; ++i) {
        const int flat = t + i * 256;
        const int r = flat >> 6, d = flat & 63;
        zlds[r * 68 + d] = z[(size_t)(row0 + r) * 64 + d];
    }
    __syncthreads();
    if (t < 16) {
        float s = 0.0f;
        for (int d = 0; d < 64; ++d) { const float v = zlds[t * 68 + d]; s += v * v; }
        zz[t] = s;
    }

    // A fragments (16x4 f32 layout: VGPR0=K{0,2}, VGPR1=K{1,3} per half-wave)
    const int arow = lane & 15;
    const int koff = (lane >> 4) << 1;
    v2f a[16];
    #pragma unroll
    for (int g = 0; g < 16; ++g)
        a[g] = *(const v2f*)&zlds[arow * 68 + g * 4 + koff];

    float m_run[8], s_run[8];
    #pragma unroll
    for (int i = 0; i < 8; ++i) { m_run[i] = NEG_BIG; s_run[i] = 0.0f; }

    const float* bp0 = wsbf + (size_t)wave * 1024 + lane * 4;
    for (int iter = 0; iter < SM_TILES_PAD / 8; ++iter) {
        const int j = iter * 8 + wave;
        const float qv = wsq[j * 16 + (lane & 15)];
        const float* bp = bp0 + (size_t)iter * 8 * 1024;
        if (iter + 1 < SM_TILES_PAD / 8)
            __builtin_prefetch(bp + 8 * 1024, 0, 1);   // next tile for this wave

        // two independent accumulation chains to halve D->C dependency depth
        v8f acc0 = {};
        v8f acc1 = {};
        #pragma unroll
        for (int p = 0; p < 8; p += 2) {
            v4f bq0 = *(const v4f*)(bp + p * 128);
            v4f bq1 = *(const v4f*)(bp + (p + 1) * 128);
            v2f b00 = {bq0.x, bq0.y};
            v2f b01 = {bq0.z, bq0.w};
            v2f b10 = {bq1.x, bq1.y};
            v2f b11 = {bq1.z, bq1.w};
            acc0 = __builtin_amdgcn_wmma_f32_16x16x4_f32(
                false, a[2 * p], false, b00, (short)0, acc0, false, false);
            acc1 = __builtin_amdgcn_wmma_f32_16x16x4_f32(
                false, a[2 * p + 2], false, b10, (short)0, acc1, false, false);
            acc0 = __builtin_amdgcn_wmma_f32_16x16x4_f32(
                false, a[2 * p + 1], false, b01, (short)0, acc0, false, false);
            acc1 = __builtin_amdgcn_wmma_f32_16x16x4_f32(
                false, a[2 * p + 3], false, b11, (short)0, acc1, false, false);
        }

        // per-lane online logsumexp (single exp per element)
        #pragma unroll
        for (int i = 0; i < 8; ++i) {
            const float val = (acc0[i] + acc1[i]) + qv;
            const float d = val - m_run[i];
            const float e = __expf(-fabsf(d));
            s_run[i] = (d <= 0.0f) ? (s_run[i] + e) : (s_run[i] * e + 1.0f);
            m_run[i] = fmaxf(m_run[i], val);
        }
    }

    // butterfly-merge (m,s) across the 16 lanes of each half (once, not per tile)
    #pragma unroll
    for (int i = 0; i < 8; ++i) {
        float m = m_run[i], s = s_run[i];
        #pragma unroll
        for (int d = 1; d < 16; d <<= 1) {
            const float om = __shfl_xor(m, d, 32);
            const float os = __shfl_xor(s, d, 32);
            const float nm = fmaxf(m, om);
            s = s * __expf(m - nm) + os * __expf(om - nm);
            m = nm;
        }
        m_run[i] = m; s_run[i] = s;
    }

    // per-wave state is replicated across each 16-lane half; one lane per half exports it
    if ((lane & 15) == 0) {
        const int half = lane >> 4;
        #pragma unroll
        for (int i = 0; i < 8; ++i) {
            red_m[wave * 16 + half * 8 + i] = m_run[i];
            red_s[wave * 16 + half * 8 + i] = s_run[i];
        }
    }
    __syncthreads();
    if (t < 16) {
        float M = NEG_BIG;
        for (int w = 0; w < 8; ++w) M = fmaxf(M, red_m[w * 16 + t]);
        float S = 0.0f;
        for (int w = 0; w < 8; ++w) S += red_s[w * 16 + t] * __expf(red_m[w * 16 + t] - M);
        rowlp[t] = M + __logf(S) + C0F - 0.5f * zz[t];
    }
    __syncthreads();
    if (t == 0) {
        float s = 0.0f;
        #pragma unroll
        for (int i = 0; i < 16; ++i) s += rowlp[i];
        partial[blockIdx.x] = s;
    }
}

// ---------------------------------------------------------------- kernel 3
__global__ void gmm_final_kernel(const float* __restrict__ partial, float* __restrict__ out) {
    __shared__ float buf[256];
    const int t = threadIdx.x;
    buf[t] = partial[t];
    __syncthreads();
    for (int s = 128; s > 0; s >>= 1) { if (t < s) buf[t] += buf[t + s]; __syncthreads(); }
    if (t == 0) out[0] = buf[0] / (float)NB - __logf((float)MW);
}

// ---------------------------------------------------------------- launch
extern "C" void kernel_launch(void* const* d_in, const int* in_sizes, int n_in,
                              void* d_out, int out_size, void* d_ws, size_t ws_size,
                              hipStream_t stream) {
    const float* zin = (const float*)d_in[0];   // [4096,1,64]
    const float* pi  = (const float*)d_in[1];   // [1,120]
    const float* mu  = (const float*)d_in[2];   // [64,15]
    float* wsf = (float*)d_ws;
    float* out = (float*)d_out;

    gmm_logpi_kernel<<<1, 128, 0, stream>>>(pi, wsf);
    gmm_prep_kernel<<<SM_TILES_PAD, 256, 0, stream>>>(mu, wsf);
    gmm_main_kernel<<<NB / 16, 256, 0, stream>>>(zin, wsf + WS_Q, wsf + WS_BFRAG, wsf + WS_PART);
    gmm_final_kernel<<<1, 256, 0, stream>>>(wsf + WS_PART, out);
}